// lightgraphconvlayer_43198781063350
// MI455X (gfx1250) — compile-verified
//
#include <hip/hip_runtime.h>
#include <stdint.h>

#define NC   50000
#define NG   3000
#define DF   64
#define NE_CG 3000000
#define NE_GC 3000000
#define NE_GG 100000

// ---------------------------------------------------------------------------
// Async gather staging: per-lane copy of 8 bytes of the source feature row
// into this wave's LDS slot via the CDNA5 async-to-LDS path (ASYNCcnt).
// ---------------------------------------------------------------------------
__device__ __forceinline__ void issue_async_row(const float* gp, unsigned lds_off) {
    unsigned long long ga = (unsigned long long)(size_t)gp;
    // s_wait_dscnt 0 guards the LDS-slot WAR (previous ds_load of this slot
    // must have retired before the async engine overwrites it).
    asm volatile("s_wait_dscnt 0\n\t"
                 "global_load_async_to_lds_b64 %0, %1, off"
                 :: "v"(lds_off), "v"(ga) : "memory");
}
__device__ __forceinline__ void issue_async_row_first(const float* gp, unsigned lds_off) {
    unsigned long long ga = (unsigned long long)(size_t)gp;
    asm volatile("global_load_async_to_lds_b64 %0, %1, off"
                 :: "v"(lds_off), "v"(ga) : "memory");
}
__device__ __forceinline__ void wait_async_le1() {
    asm volatile("s_wait_asynccnt 1" ::: "memory");
}
__device__ __forceinline__ void wait_async_le0() {
    asm volatile("s_wait_asynccnt 0" ::: "memory");
}

__device__ __forceinline__ void consume_edge(const float* slot, int d, float w,
                                             float* aggR, int lane) {
    float v0 = slot[0];
    float v1 = slot[1];
    float* outp = aggR + (size_t)d * DF + (size_t)(lane * 2);
    unsafeAtomicAdd(outp,     v0 * w);   // global_atomic_add_f32, no return
    unsafeAtomicAdd(outp + 1, v1 * w);
}

// One wave per edge; depth-2 async double-buffer of the gathered 256B row.
// repMask = (replica_count - 1), replica_count a power of two.
__global__ __launch_bounds__(256) void edge_scatter_kernel(
    const float* __restrict__ feat, const float* __restrict__ wj,
    const int* __restrict__ src, const int* __restrict__ dst,
    float* __restrict__ agg, int nEdges, unsigned repMask, int ndst)
{
    __shared__ __align__(16) float stage[8][2][DF];
    const int lane = (int)(threadIdx.x & 31u);
    const int wib  = (int)(threadIdx.x >> 5u);
    const long long nwaves = (long long)gridDim.x * 8;
    long long e = (long long)blockIdx.x * 8 + wib;

    float* aggR = agg + (size_t)(blockIdx.x & repMask) * (size_t)ndst * DF;

    float* s0 = &stage[wib][0][lane * 2];
    float* s1 = &stage[wib][1][lane * 2];
    const unsigned lds0 = (unsigned)(size_t)s0;   // low 32 bits = LDS offset
    const unsigned lds1 = (unsigned)(size_t)s1;

    int   d0 = 0, d1 = 0;
    float w0 = 0.f, w1 = 0.f;

    if (e < nEdges) {   // wave-uniform: EXEC all-ones for async issue
        int s = __builtin_nontemporal_load(src + e);
        d0 = __builtin_nontemporal_load(dst + e);
        w0 = wj[s];
        issue_async_row_first(feat + (size_t)s * DF + (size_t)(lane * 2), lds0);
    }
    while (e < nEdges) {
        long long e1 = e + nwaves;
        if (e1 < nEdges) {
            int s = __builtin_nontemporal_load(src + e1);
            d1 = __builtin_nontemporal_load(dst + e1);
            w1 = wj[s];
            issue_async_row(feat + (size_t)s * DF + (size_t)(lane * 2), lds1);
            wait_async_le1();                 // slot0 complete (in-order)
        } else {
            wait_async_le0();
        }
        consume_edge(s0, d0, w0, aggR, lane);
        if (e1 >= nEdges) break;

        long long e2 = e1 + nwaves;
        if (e2 < nEdges) {
            int s = __builtin_nontemporal_load(src + e2);
            d0 = __builtin_nontemporal_load(dst + e2);
            w0 = wj[s];
            issue_async_row(feat + (size_t)s * DF + (size_t)(lane * 2), lds0);
            wait_async_le1();                 // slot1 complete
        } else {
            wait_async_le0();
        }
        consume_edge(s1, d1, w1, aggR, lane);
        e = e2;
    }
}

// Per-source-node weights wj = cj * mask (mask already holds 1/KEEP scaling).
__global__ __launch_bounds__(256) void init_wj_kernel(
    const float* __restrict__ cj_cell, const float* __restrict__ mask_exp,
    const float* __restrict__ cj_gene, const float* __restrict__ mask_rev,
    const float* __restrict__ cjj_gene, const float* __restrict__ mask_gg,
    float* __restrict__ wj_cell, float* __restrict__ wj_rev, float* __restrict__ wj_gg)
{
    int i = blockIdx.x * blockDim.x + threadIdx.x;
    if (i < NC) wj_cell[i] = cj_cell[i] * mask_exp[i];
    if (i < NG) {
        wj_rev[i] = cj_gene[i] * mask_rev[i];
        wj_gg[i]  = cjj_gene[i] * mask_gg[i];
    }
}

__global__ __launch_bounds__(256) void finalize_cells_kernel(
    const float* __restrict__ agg_c, const float* __restrict__ ci_cell,
    float* __restrict__ c_out)
{
    int idx = blockIdx.x * blockDim.x + threadIdx.x;   // over NC*DF/4
    if (idx < NC * (DF / 4)) {
        int node = idx / (DF / 4);
        float4 v = ((const float4*)agg_c)[idx];
        float ci = ci_cell[node];
        float4 r; r.x = v.x * ci; r.y = v.y * ci; r.z = v.z * ci; r.w = v.w * ci;
        ((float4*)c_out)[idx] = r;
    }
}

__global__ __launch_bounds__(256) void finalize_genes_kernel(
    const float* __restrict__ agg_gexp, const float* __restrict__ agg_ggg,
    const float* __restrict__ ci_gene, const float* __restrict__ cii_gene,
    float* __restrict__ g_out, int nrep)
{
    int idx = blockIdx.x * blockDim.x + threadIdx.x;   // over NG*DF/4
    if (idx < NG * (DF / 4)) {
        int node = idx / (DF / 4);
        float4 s; s.x = s.y = s.z = s.w = 0.f;
        for (int r = 0; r < nrep; ++r) {
            float4 v = ((const float4*)(agg_gexp + (size_t)r * NG * DF))[idx];
            s.x += v.x; s.y += v.y; s.z += v.z; s.w += v.w;
        }
        float4 gg = ((const float4*)agg_ggg)[idx];
        float a = 0.5f * ci_gene[node];
        float b = 0.5f * cii_gene[node];
        float4 r; r.x = a * s.x + b * gg.x; r.y = a * s.y + b * gg.y;
        r.z = a * s.z + b * gg.z; r.w = a * s.w + b * gg.w;
        ((float4*)g_out)[idx] = r;
    }
}

static inline int blocks_for_edges(int nEdges) {
    long long waves = (nEdges + 63) / 64;      // >= ~64 edges per wave
    long long b = (waves + 7) / 8;
    if (b < 1) b = 1;
    if (b > 2048) b = 2048;
    return (int)b;
}

extern "C" void kernel_launch(void* const* d_in, const int* in_sizes, int n_in,
                              void* d_out, int out_size, void* d_ws, size_t ws_size,
                              hipStream_t stream)
{
    (void)in_sizes; (void)n_in; (void)out_size;
    const float* c_feat   = (const float*)d_in[0];
    const float* g_feat   = (const float*)d_in[1];
    const float* cj_cell  = (const float*)d_in[2];
    const float* ci_cell  = (const float*)d_in[3];
    const float* cj_gene  = (const float*)d_in[4];
    const float* ci_gene  = (const float*)d_in[5];
    const float* cjj_gene = (const float*)d_in[6];
    const float* cii_gene = (const float*)d_in[7];
    const float* mask_exp = (const float*)d_in[8];
    const float* mask_rev = (const float*)d_in[9];
    const float* mask_gg  = (const float*)d_in[10];
    const int* src_cg = (const int*)d_in[11];
    const int* dst_cg = (const int*)d_in[12];
    const int* src_gc = (const int*)d_in[13];
    const int* dst_gc = (const int*)d_in[14];
    const int* src_gg = (const int*)d_in[15];
    const int* dst_gg = (const int*)d_in[16];

    // Replicate the hot gene-exp accumulator (3M edges -> 3000 rows) 8x to
    // cut atomic contention, if scratch allows. rep is a power of two.
    size_t need8 = (size_t)(NC + 2 * NG) * 4
                 + ((size_t)NC * DF + (size_t)8 * NG * DF + (size_t)NG * DF) * 4;
    int rep = (ws_size >= need8) ? 8 : 1;

    float* ws       = (float*)d_ws;
    float* wj_cell  = ws;
    float* wj_rev   = wj_cell + NC;
    float* wj_gg    = wj_rev + NG;
    float* agg_c    = wj_gg + NG;
    float* agg_gexp = agg_c + (size_t)NC * DF;
    float* agg_ggg  = agg_gexp + (size_t)rep * NG * DF;

    size_t aggFloats = (size_t)NC * DF + (size_t)rep * NG * DF + (size_t)NG * DF;
    hipMemsetAsync(agg_c, 0, aggFloats * sizeof(float), stream);

    init_wj_kernel<<<(NC + 255) / 256, 256, 0, stream>>>(
        cj_cell, mask_exp, cj_gene, mask_rev, cjj_gene, mask_gg,
        wj_cell, wj_rev, wj_gg);

    // cell -> gene ('exp'): replicated accumulator
    edge_scatter_kernel<<<blocks_for_edges(NE_CG), 256, 0, stream>>>(
        c_feat, wj_cell, src_cg, dst_cg, agg_gexp, NE_CG, (unsigned)(rep - 1), NG);
    // gene -> cell ('reverse-exp'): low contention, single accumulator
    edge_scatter_kernel<<<blocks_for_edges(NE_GC), 256, 0, stream>>>(
        g_feat, wj_rev, src_gc, dst_gc, agg_c, NE_GC, 0u, NC);
    // gene -> gene ('co-exp')
    edge_scatter_kernel<<<blocks_for_edges(NE_GG), 256, 0, stream>>>(
        g_feat, wj_gg, src_gg, dst_gg, agg_ggg, NE_GG, 0u, NG);

    float* c_out = (float*)d_out;                       // [NC, 64]
    float* g_out = (float*)d_out + (size_t)NC * DF;     // [NG, 64]
    finalize_cells_kernel<<<(NC * (DF / 4) + 255) / 256, 256, 0, stream>>>(
        agg_c, ci_cell, c_out);
    finalize_genes_kernel<<<(NG * (DF / 4) + 255) / 256, 256, 0, stream>>>(
        agg_gexp, agg_ggg, ci_gene, cii_gene, g_out, rep);
}